// YOLOV0Loss_10754598109418
// MI455X (gfx1250) — compile-verified
//
#include <hip/hip_runtime.h>

// YOLO-v0 loss: sum over cells of
//   obj:   5*|coord_pred - coord_gt|^2 + (conf_pred - 1)^2   if conf_gt > 0
//   noobj: 0.5*conf_pred^2                                    otherwise
// divided by batch B=128.  Inputs: [B,W,H,5] f32, channel = (conf,x,y,w,h).
//
// Bandwidth-bound: 335.5 MB read -> ~14.4us floor at 23.3 TB/s.
// Phase 1: vectorized float4 loads, per-thread f32 accumulation, block reduce -> d_ws.
// Phase 2: one wave32 folds block partials via V_WMMA_F32_16X16X4_F32 (B = ones).

typedef __attribute__((ext_vector_type(2))) float v2f;
typedef __attribute__((ext_vector_type(8))) float v8f;

#define LAMBDA_COOR  5.0f
#define LAMBDA_NOOBJ 0.5f
#define NBLOCKS  2048
#define NTHREADS 256

__device__ __forceinline__ float cell_loss(float cp, float cg,
                                           float dx, float dy, float dw, float dh) {
    float coord = dx * dx + dy * dy + dw * dw + dh * dh;
    float cm1   = cp - 1.0f;
    float obj   = LAMBDA_COOR * coord + cm1 * cm1;
    float noobj = LAMBDA_NOOBJ * cp * cp;
    return (cg > 0.0f) ? obj : noobj;
}

__global__ __launch_bounds__(NTHREADS)
void yolo_partial_kernel(const float* __restrict__ outp,
                         const float* __restrict__ tgtp,
                         float* __restrict__ partials,
                         int nUnits, long long nCells) {
    const float4* __restrict__ out4 = (const float4*)outp;
    const float4* __restrict__ tgt4 = (const float4*)tgtp;

    int tid = blockIdx.x * NTHREADS + threadIdx.x;
    const int stride = NBLOCKS * NTHREADS;
    float acc = 0.0f;

    // Each unit = 4 cells = 20 floats = 5 aligned float4 per tensor.
    for (int u = tid; u < nUnits; u += stride) {
        size_t base = (size_t)u * 5;
        float po[20], pt[20];
#pragma unroll
        for (int j = 0; j < 5; ++j) {
            float4 a = out4[base + j];
            float4 b = tgt4[base + j];
            po[4 * j + 0] = a.x; po[4 * j + 1] = a.y;
            po[4 * j + 2] = a.z; po[4 * j + 3] = a.w;
            pt[4 * j + 0] = b.x; pt[4 * j + 1] = b.y;
            pt[4 * j + 2] = b.z; pt[4 * j + 3] = b.w;
        }
#pragma unroll
        for (int c = 0; c < 4; ++c) {
            int k = 5 * c;
            acc += cell_loss(po[k], pt[k],
                             po[k + 1] - pt[k + 1], po[k + 2] - pt[k + 2],
                             po[k + 3] - pt[k + 3], po[k + 4] - pt[k + 4]);
        }
    }

    // Tail cells (if nCells % 4 != 0) — no-op for the harness shape.
    if (tid == 0) {
        for (long long cell = (long long)nUnits * 4; cell < nCells; ++cell) {
            size_t k = (size_t)cell * 5;
            acc += cell_loss(outp[k], tgtp[k],
                             outp[k + 1] - tgtp[k + 1], outp[k + 2] - tgtp[k + 2],
                             outp[k + 3] - tgtp[k + 3], outp[k + 4] - tgtp[k + 4]);
        }
    }

    // Wave32 reduction, then cross-wave via LDS.
#pragma unroll
    for (int off = 16; off > 0; off >>= 1) acc += __shfl_xor(acc, off, 32);

    __shared__ float lds[NTHREADS / 32];
    int lane = threadIdx.x & 31;
    int wid  = threadIdx.x >> 5;
    if (lane == 0) lds[wid] = acc;
    __syncthreads();
    if (threadIdx.x < NTHREADS / 32) {
        float v = lds[threadIdx.x];
#pragma unroll
        for (int off = (NTHREADS / 32) / 2; off > 0; off >>= 1)
            v += __shfl_xor(v, off, 32);
        if (threadIdx.x == 0) partials[blockIdx.x] = v;
    }
}

// One wave. Sums NBLOCKS partials with V_WMMA_F32_16X16X4_F32 against an all-ones
// B matrix: D[m][n] = sum_k A[m][k] + C[m][n], so C accumulates row sums; the total
// is the sum of column N=0 of C. EXEC is all-1s (no divergence before the WMMAs).
__global__ __launch_bounds__(32)
void yolo_finalize_kernel(const float* __restrict__ partials,
                          float* __restrict__ out, float invBatch) {
    int lane = threadIdx.x;

    v8f c = {};
    v2f bones; bones[0] = 1.0f; bones[1] = 1.0f;

    for (int base = 0; base < NBLOCKS; base += 64) {
        v2f a;
        a[0] = partials[base + lane];
        a[1] = partials[base + 32 + lane];
        // (neg_a, A, neg_b, B, c_mod, C, reuse_a, reuse_b)
        c = __builtin_amdgcn_wmma_f32_16x16x4_f32(false, a, false, bones,
                                                  (short)0, c, false, false);
    }

    // C/D layout: VGPR r, lanes 0-15 -> C[M=r][N=lane]; lanes 16-31 -> C[M=r+8][N=lane-16].
    // Lane 0 sums C[0..7][0], lane 16 sums C[8..15][0]; together = full sum.
    float s = c[0] + c[1] + c[2] + c[3] + c[4] + c[5] + c[6] + c[7];
    float s_hi = __shfl(s, 16, 32);
    if (lane == 0) out[0] = (s + s_hi) * invBatch;
}

extern "C" void kernel_launch(void* const* d_in, const int* in_sizes, int n_in,
                              void* d_out, int out_size, void* d_ws, size_t ws_size,
                              hipStream_t stream) {
    const float* outputs = (const float*)d_in[0];
    const float* targets = (const float*)d_in[1];
    float* out      = (float*)d_out;
    float* partials = (float*)d_ws;   // NBLOCKS floats = 8 KB

    long long nElem  = (long long)in_sizes[0];  // B*W*H*5
    long long nCells = nElem / 5;               // 8,388,608
    int nUnits = (int)(nCells / 4);             // 2,097,152

    // Reference divides by outputs.shape[0] = 128 (fixed by the harness setup).
    const float invBatch = 1.0f / 128.0f;

    yolo_partial_kernel<<<NBLOCKS, NTHREADS, 0, stream>>>(outputs, targets,
                                                          partials, nUnits, nCells);
    yolo_finalize_kernel<<<1, 32, 0, stream>>>(partials, out, invBatch);
}